// UNIPoint_14164802142739
// MI455X (gfx1250) — compile-verified
//
#include <hip/hip_runtime.h>
#include <math.h>

#define BATCH 256
#define SEQ   2048
#define FEAT  64
#define NROWS (BATCH * SEQ)   // 524288 flattened (b,t) rows; row r = b*SEQ + t

typedef float v2f __attribute__((ext_vector_type(2)));
typedef float v8f __attribute__((ext_vector_type(8)));

// ---------------------------------------------------------------------------
// Kernel A: proj[t*B + b] = X[b,t,:] . w_ih + (b_ih + b_hh), via
// V_WMMA_F32_16X16X4_F32. One wave handles a tile of 16 consecutive rows
// (all same b, consecutive t, since SEQ % 16 == 0). A-fragment layout
// (32-bit A 16x4): lanes 0-15 = rows M=0..15 holding K={k,k+1}; lanes 16-31 =
// same rows holding K={k+2,k+3}. B is 4x16 with w in column N=0 only, so
// D[:,0] (lanes 0 and 16, VGPRs 0..7) accumulates the dot products; C is
// preloaded with the bias in column 0.
//
// Structure: (a) build all 16 B fragments in registers with unconditional LDS
// broadcast reads + v_cndmask masking (no EXEC save/restore in the loop),
// (b) load the full 4KB A tile into registers so all 16 global_load_b64 are
// in flight simultaneously, (c) 16 back-to-back WMMAs on the accumulator.
// ---------------------------------------------------------------------------
__global__ void __launch_bounds__(256)
proj_kernel(const float* __restrict__ X,
            const float* __restrict__ W_ih,
            const float* __restrict__ b_ih,
            const float* __restrict__ b_hh,
            float* __restrict__ proj,
            float* __restrict__ tvals) {
    __shared__ float sW[FEAT];
    const int tid = threadIdx.x;
    if (tid < FEAT) sW[tid] = W_ih[tid];
    __syncthreads();

    const int lane = tid & 31;
    const int wave = tid >> 5;
    const int m    = lane & 15;   // row within tile
    const int hi   = lane >> 4;   // 0: K={k,k+1}, 1: K={k+2,k+3}
    const long tile   = (long)blockIdx.x * 8 + wave;
    const long r_base = tile * 16;
    const long row    = r_base + m;
    const float* xrow = X + row * FEAT + hi * 2;

    const float bias = b_ih[0] + b_hh[0];
    const bool colZeroLane = (m == 0);   // lanes 0 and 16 own D[:,0]

    // ---- B fragments: unconditional LDS broadcast + value select ----
    v2f bf[16];
#pragma unroll
    for (int kk = 0; kk < 16; ++kk) {
        const int k = kk * 4 + hi * 2;
        const float wx = sW[k];          // all lanes read (broadcast)
        const float wy = sW[k + 1];
        bf[kk].x = colZeroLane ? wx : 0.0f;   // v_cndmask, no branch
        bf[kk].y = colZeroLane ? wy : 0.0f;
    }

    // ---- A fragments: stream the whole 4KB tile into registers ----
    v2f af[16];
#pragma unroll
    for (int kk = 0; kk < 16; ++kk)
        af[kk] = *(const v2f*)(xrow + kk * 4);   // 8B-aligned

    // ---- Accumulator: bias in column 0 ----
    v8f c;
#pragma unroll
    for (int i = 0; i < 8; ++i) c[i] = colZeroLane ? bias : 0.0f;

    // ---- 16 back-to-back WMMAs ----
#pragma unroll
    for (int kk = 0; kk < 16; ++kk)
        c = __builtin_amdgcn_wmma_f32_16x16x4_f32(
                /*neg_a=*/false, af[kk], /*neg_b=*/false, bf[kk],
                /*c_mod=*/(short)0, c, /*reuse_a=*/false, /*reuse_b=*/false);

    // ---- t_t = X[b,t,1] sits in af[0].y on hi==0 lanes ----
    if (hi == 0) {
        const int bidx = (int)(row >> 11);        // / SEQ
        const int t    = (int)(row & (SEQ - 1));
        tvals[t * BATCH + bidx] = af[0].y;
    }

    // ---- Store D[:,0]: lane 0 holds M=0..7, lane 16 holds M=8..15 ----
    if (colZeroLane) {
        const long rr    = r_base + hi * 8;
        const int  bidx  = (int)(rr >> 11);
        const int  tbase = (int)(rr & (SEQ - 1));
#pragma unroll
        for (int i = 0; i < 8; ++i)
            proj[(tbase + i) * BATCH + bidx] = c[i];
    }
}

// ---------------------------------------------------------------------------
// Kernel B: the serial scan. 256 independent chains (one per batch element),
// 8 wave32s. Critical path per step = v_fma_f32 + v_tanh_f32 (hardware tanh
// on gfx1250). proj is in [t][B] layout so each wave load is one coalesced
// 128B line; unroll-8 lets the loads run ahead of the dependency chain.
// ---------------------------------------------------------------------------
static __device__ __forceinline__ float fast_tanh(float x) {
#if __has_builtin(__builtin_amdgcn_tanhf)
    return __builtin_amdgcn_tanhf(x);
#elif __has_builtin(__builtin_amdgcn_tanh_f32)
    return __builtin_amdgcn_tanh_f32(x);
#else
    return tanhf(x);
#endif
}

__global__ void __launch_bounds__(32)
scan_kernel(const float* __restrict__ proj,
            const float* __restrict__ hx0,
            const float* __restrict__ W_hh,
            float* __restrict__ hidden) {
    const int b = blockIdx.x * 32 + threadIdx.x;   // 0..255
    float h = hx0[b];
    const float w = W_hh[0];
    const float* p = proj + b;
    float* o = hidden + b;
#pragma unroll 8
    for (int t = 0; t < SEQ; ++t) {
        const float a = p[(long)t * BATCH];        // bias already folded in
        h = fast_tanh(fmaf(h, w, a));
        o[(long)t * BATCH] = h;
    }
}

// ---------------------------------------------------------------------------
// Kernel C: intensity head, fully parallel over all (t,b).
// s = sum_j relu((W[2j]*h + b[2j]) * t_t + (W[2j+1]*h + b[2j+1]))
// lam = softplus(s) = max(s,0) + log1p(exp(-|s|))   (stable form)
// ---------------------------------------------------------------------------
__global__ void __launch_bounds__(256)
lam_kernel(const float* __restrict__ hidden,
           const float* __restrict__ tvals,
           const float* __restrict__ W_h2p,
           const float* __restrict__ b_h2p,
           float* __restrict__ lam_out) {
    __shared__ float sW[64], sb[64];
    const int tid = threadIdx.x;
    if (tid < 64) { sW[tid] = W_h2p[tid]; sb[tid] = b_h2p[tid]; }
    __syncthreads();

    const long idx = (long)blockIdx.x * 256 + tid;   // t*BATCH + b
    const float h  = hidden[idx];
    const float tt = tvals[idx];

    float s = 0.0f;
#pragma unroll
    for (int j = 0; j < 32; ++j) {
        const float alpha = fmaf(h, sW[2 * j],     sb[2 * j]);
        const float beta  = fmaf(h, sW[2 * j + 1], sb[2 * j + 1]);
        const float v     = fmaf(alpha, tt, beta);
        s += fmaxf(v, 0.0f);
    }
    const float lam = fmaxf(s, 0.0f) + log1pf(__expf(-fabsf(s)));
    lam_out[idx] = lam;
}

// ---------------------------------------------------------------------------
extern "C" void kernel_launch(void* const* d_in, const int* in_sizes, int n_in,
                              void* d_out, int out_size, void* d_ws, size_t ws_size,
                              hipStream_t stream) {
    const float* X     = (const float*)d_in[0];
    const float* hx0   = (const float*)d_in[1];
    const float* W_ih  = (const float*)d_in[2];
    const float* b_ih  = (const float*)d_in[3];
    const float* W_hh  = (const float*)d_in[4];
    const float* b_hh  = (const float*)d_in[5];
    const float* W_h2p = (const float*)d_in[6];
    const float* b_h2p = (const float*)d_in[7];

    float* proj  = (float*)d_ws;          // NROWS floats, [t][B] layout (2 MB)
    float* tvals = proj + NROWS;          // NROWS floats, [t][B] layout (2 MB)

    float* hidden = (float*)d_out;        // (S,B,1) -> [t*B + b]
    float* lam    = hidden + NROWS;       // (S,B)   -> [t*B + b]

    // A: 32768 16-row tiles, 8 waves (tiles) per 256-thread block
    proj_kernel<<<NROWS / 16 / 8, 256, 0, stream>>>(X, W_ih, b_ih, b_hh, proj, tvals);
    // B: one wave per block, 8 blocks -> 256 chains spread across WGPs
    scan_kernel<<<8, 32, 0, stream>>>(proj, hx0, W_hh, hidden);
    // C: one thread per (t,b)
    lam_kernel<<<NROWS / 256, 256, 0, stream>>>(hidden, tvals, W_h2p, b_h2p, lam);
}